// PYG_GMM_50697793962353
// MI455X (gfx1250) — compile-verified
//
#include <hip/hip_runtime.h>

typedef float v2f __attribute__((ext_vector_type(2)));
typedef float v8f __attribute__((ext_vector_type(8)));

#define EPSQ 1e-15f

// ---------------------------------------------------------------------------
// Zero fill (grid-stride)
// ---------------------------------------------------------------------------
__global__ void zero_kernel(float* __restrict__ p, long n) {
  long i = (long)blockIdx.x * blockDim.x + threadIdx.x;
  long stride = (long)gridDim.x * blockDim.x;
  for (; i < n; i += stride) p[i] = 0.0f;
}

// ---------------------------------------------------------------------------
// Per-edge Gaussian weights for BOTH layers (single pass over edge_attr)
// plus in-degree accumulation (degree identical across layers).
// ---------------------------------------------------------------------------
__global__ void gauss_kernel(const int* __restrict__ ei,   // [2][E]
                             const float* __restrict__ ea, // [E][8]
                             const float* __restrict__ mu1, const float* __restrict__ sg1,
                             const float* __restrict__ mu2, const float* __restrict__ sg2,
                             float* __restrict__ w1, float* __restrict__ w2,
                             float* __restrict__ cnt, int E) {
  int e = blockIdx.x * blockDim.x + threadIdx.x;
  if (e >= E) return;
  float s1 = 0.0f, s2 = 0.0f;
#pragma unroll
  for (int d = 0; d < 8; ++d) {
    float a  = ea[(long)e * 8 + d];
    float d1 = a - mu1[d];
    float d2 = a - mu2[d];
    float q1 = sg1[d], q2 = sg2[d];
    s1 += (d1 * d1) / (EPSQ + q1 * q1);
    s2 += (d2 * d2) / (EPSQ + q2 * q2);
  }
  w1[e] = __expf(-0.5f * s1);
  w2[e] = __expf(-0.5f * s2);
  int dst = ei[E + e];
  atomicAdd(&cnt[dst], 1.0f);
}

// ---------------------------------------------------------------------------
// Dense transform via V_WMMA_F32_16X16X4_F32:
//   XG   = X @ G          (N x 32) @ (32 x 32)
//   BASE = X @ R + bias
// One wave -> 16-row strip, 4 accumulators (2 col-tiles x {G,R}).
// A frag (16x4 f32): lane&15 = M; VGPR0/1 hold K = k+2*(lane>=16), +1.
// B frag (4x16 f32): mirrored (lane&15 = N).
// C/D (16x16 f32):   VGPR r -> row r + 8*(lane>=16), col = lane&15.
// ---------------------------------------------------------------------------
__global__ void __launch_bounds__(256)
gemm_kernel(const float* __restrict__ X, const float* __restrict__ G,
            const float* __restrict__ R, const float* __restrict__ B,
            float* __restrict__ XG, float* __restrict__ BASE,
            int N, int ntiles) {
  int wave = blockIdx.x * (blockDim.x >> 5) + (threadIdx.x >> 5);
  if (wave >= ntiles) return;
  int lane = threadIdx.x & 31;
  int half = lane >> 4;
  int m    = lane & 15;
  int row0 = wave * 16;
  int arow = row0 + m;
  if (arow >= N) arow = N - 1;  // clamp; garbage rows discarded at store

  v8f accg0 = {}, accg1 = {}, accr0 = {}, accr1 = {};

#pragma unroll
  for (int k = 0; k < 32; k += 4) {
    int kk = k + half * 2;
    v2f a = *(const v2f*)(X + (long)arow * 32 + kk);
    v2f bg0, bg1, br0, br1;
    bg0.x = G[kk * 32 + m];        bg0.y = G[(kk + 1) * 32 + m];
    bg1.x = G[kk * 32 + 16 + m];   bg1.y = G[(kk + 1) * 32 + 16 + m];
    br0.x = R[kk * 32 + m];        br0.y = R[(kk + 1) * 32 + m];
    br1.x = R[kk * 32 + 16 + m];   br1.y = R[(kk + 1) * 32 + 16 + m];
    accg0 = __builtin_amdgcn_wmma_f32_16x16x4_f32(false, a, false, bg0, (short)0, accg0, false, false);
    accg1 = __builtin_amdgcn_wmma_f32_16x16x4_f32(false, a, false, bg1, (short)0, accg1, false, false);
    accr0 = __builtin_amdgcn_wmma_f32_16x16x4_f32(false, a, false, br0, (short)0, accr0, false, false);
    accr1 = __builtin_amdgcn_wmma_f32_16x16x4_f32(false, a, false, br1, (short)0, accr1, false, false);
  }

  float b0 = B[m], b1 = B[16 + m];
#pragma unroll
  for (int r = 0; r < 8; ++r) {
    int row = row0 + half * 8 + r;
    if (row < N) {
      XG[(long)row * 32 + m]          = accg0[r];
      XG[(long)row * 32 + 16 + m]     = accg1[r];
      BASE[(long)row * 32 + m]        = accr0[r] + b0;
      BASE[(long)row * 32 + 16 + m]   = accr1[r] + b1;
    }
  }
}

// ---------------------------------------------------------------------------
// Edge scatter: agg[dst] += xg[src] * w[e].  8 lanes per edge, float4 each.
// xg (12.8 MB) and agg (12.8 MB) are L2-resident (192 MB L2).
// ---------------------------------------------------------------------------
__global__ void edge_kernel(const int* __restrict__ ei, const float* __restrict__ w,
                            const float* __restrict__ XG, float* __restrict__ AGG,
                            int E) {
  long t = (long)blockIdx.x * blockDim.x + threadIdx.x;
  long e = t >> 3;
  if (e >= E) return;
  int q = (int)(t & 7);
  int s = ei[e];
  int d = ei[E + e];
  float ww = w[e];
  float4 v = *(const float4*)(XG + (long)s * 32 + q * 4);
  float* base = AGG + (long)d * 32 + q * 4;
  atomicAdd(base + 0, v.x * ww);
  atomicAdd(base + 1, v.y * ww);
  atomicAdd(base + 2, v.z * ww);
  atomicAdd(base + 3, v.w * ww);
}

// ---------------------------------------------------------------------------
// Finish: out = agg / max(cnt,1) + base
// ---------------------------------------------------------------------------
__global__ void finish_kernel(const float* __restrict__ AGG, const float* __restrict__ BASE,
                              const float* __restrict__ CNT, float* __restrict__ OUT, int N) {
  long i = (long)blockIdx.x * blockDim.x + threadIdx.x;
  if (i >= (long)N * 32) return;
  int n = (int)(i >> 5);
  float c = CNT[n];
  c = c > 1.0f ? c : 1.0f;
  OUT[i] = AGG[i] / c + BASE[i];
}

// ---------------------------------------------------------------------------
extern "C" void kernel_launch(void* const* d_in, const int* in_sizes, int n_in,
                              void* d_out, int out_size, void* d_ws, size_t ws_size,
                              hipStream_t stream) {
  const int*   ei  = (const int*)d_in[0];
  const float* ea  = (const float*)d_in[1];
  const float* x   = (const float*)d_in[2];
  const float* g1  = (const float*)d_in[3];
  const float* mu1 = (const float*)d_in[4];
  const float* sg1 = (const float*)d_in[5];
  const float* r1  = (const float*)d_in[6];
  const float* b1  = (const float*)d_in[7];
  const float* g2  = (const float*)d_in[8];
  const float* mu2 = (const float*)d_in[9];
  const float* sg2 = (const float*)d_in[10];
  const float* r2  = (const float*)d_in[11];
  const float* b2  = (const float*)d_in[12];

  const int E = in_sizes[0] / 2;
  const int N = in_sizes[2] / 32;

  float* ws   = (float*)d_ws;
  float* w1   = ws;
  float* w2   = w1 + E;
  float* cnt  = w2 + E;
  float* xg   = cnt + N;
  float* base = xg + (long)N * 32;
  float* agg  = base + (long)N * 32;
  float* h    = agg + (long)N * 32;
  float* out  = (float*)d_out;

  const int ntiles     = (N + 15) / 16;
  const int gemmBlocks = (ntiles + 7) / 8;            // 8 waves / 256-thread block
  const int edgeBlocks = (int)(((long)E * 8 + 255) / 256);
  const int finBlocks  = (int)(((long)N * 32 + 255) / 256);
  const int gBlocks    = (E + 255) / 256;

  // --- shared pre-pass ---
  zero_kernel<<<2048, 256, 0, stream>>>(cnt, N);
  zero_kernel<<<2048, 256, 0, stream>>>(agg, (long)N * 32);
  gauss_kernel<<<gBlocks, 256, 0, stream>>>(ei, ea, mu1, sg1, mu2, sg2, w1, w2, cnt, E);

  // --- layer 1 ---
  gemm_kernel<<<gemmBlocks, 256, 0, stream>>>(x, g1, r1, b1, xg, base, N, ntiles);
  edge_kernel<<<edgeBlocks, 256, 0, stream>>>(ei, w1, xg, agg, E);
  finish_kernel<<<finBlocks, 256, 0, stream>>>(agg, base, cnt, h, N);

  // --- layer 2 ---
  zero_kernel<<<2048, 256, 0, stream>>>(agg, (long)N * 32);
  gemm_kernel<<<gemmBlocks, 256, 0, stream>>>(h, g2, r2, b2, xg, base, N, ntiles);
  edge_kernel<<<edgeBlocks, 256, 0, stream>>>(ei, w2, xg, agg, E);
  finish_kernel<<<finBlocks, 256, 0, stream>>>(agg, base, cnt, out, N);
}